// Net_kv125_74637941669907
// MI455X (gfx1250) — compile-verified
//
#include <hip/hip_runtime.h>

// ---------------------------------------------------------------------------
// CDNA5 (gfx1250) sparse submanifold conv net:
//   7 levels x (two 5^3 SubMConv gather-GEMMs) + sparse 2x max pool.
// GEMMs use V_WMMA_F32_16X16X4_F32 (exact f32, matches f32 reference).
// ---------------------------------------------------------------------------

typedef __attribute__((ext_vector_type(2))) float v2f;
typedef __attribute__((ext_vector_type(8))) float v8f;

#define KTAPS 125

// One wave32 per block. Each wave computes RT row-tiles (RT*16 voxels) x
// COT co-tiles (COT*16 output channels), accumulating over 125 taps x CIN
// with f32 WMMA (K=4). A fragments are gathered through the neighbor table
// and reused across COT co-tiles; B fragments are weight slices (L2-resident,
// reused RT x across row tiles, prefetched one tap ahead).
template <int CIN, int CO, int RT, int COT>
__global__ __launch_bounds__(32) void subm_conv_wmma(
    const float* __restrict__ x,     // [N, CIN]
    const int* __restrict__ nbr,     // [N, 125]
    const float* __restrict__ w,     // [125, CIN, CO]
    float* __restrict__ out,         // [N, CO]
    int N) {
  const int lane = threadIdx.x;        // 0..31 (wave32)
  const int m    = lane & 15;          // A row / B,D column within tile
  const int kh   = (lane >> 4) << 1;   // K sub-offset: 0 (lo half) / 2 (hi half)
  const int co0  = blockIdx.y * (16 * COT);
  const int row0 = blockIdx.x * (16 * RT);

  v8f acc[RT][COT] = {};

  int  nbase[RT];
  bool rowok[RT];
#pragma unroll
  for (int t = 0; t < RT; ++t) {
    const int row = row0 + t * 16 + m;
    rowok[t] = row < N;
    nbase[t] = row * KTAPS;
  }

  for (int k = 0; k < KTAPS; ++k) {
    const float* wk = w + (size_t)k * (CIN * CO);

    // Prefetch next tap's B slice (predictable stream; global_prefetch_b8).
    if (k + 1 < KTAPS && lane < CIN)
      __builtin_prefetch(wk + (size_t)(CIN * CO) + lane * CO + co0, 0, 1);

    // Gather neighbor row ids for this tap; clamp invalid rows to row 0 and
    // zero their A fragments with a post-load select.
    const float* xrow[RT];
    bool aok[RT];
#pragma unroll
    for (int t = 0; t < RT; ++t) {
      const int r = rowok[t] ? nbr[nbase[t] + k] : -1;
      aok[t]  = r >= 0;
      xrow[t] = x + (size_t)(aok[t] ? r : 0) * CIN;
    }

#pragma unroll 4
    for (int ci = 0; ci < CIN; ci += 4) {
      const int c0 = ci + kh;

      // B fragments: w[k, c0, col], w[k, c0+1, col] for each co tile.
      v2f b[COT];
#pragma unroll
      for (int q = 0; q < COT; ++q) {
        const int col = co0 + q * 16 + m;
        if constexpr (CIN % 4 == 0) {
          b[q].x = wk[c0 * CO + col];
          b[q].y = wk[(c0 + 1) * CO + col];
        } else {
          b[q].x = (c0     < CIN) ? wk[c0 * CO + col]       : 0.0f;
          b[q].y = (c0 + 1 < CIN) ? wk[(c0 + 1) * CO + col] : 0.0f;
        }
      }

#pragma unroll
      for (int t = 0; t < RT; ++t) {
        v2f a;
        if constexpr (CIN % 4 == 0) {
          // Adjacent + 8B aligned: single b64 load, then lane-select zero.
          const v2f av = *(const v2f*)(xrow[t] + c0);
          a.x = aok[t] ? av.x : 0.0f;
          a.y = aok[t] ? av.y : 0.0f;
        } else {
          a.x = (aok[t] && c0     < CIN) ? xrow[t][c0]     : 0.0f;
          a.y = (aok[t] && c0 + 1 < CIN) ? xrow[t][c0 + 1] : 0.0f;
        }
#pragma unroll
        for (int q = 0; q < COT; ++q)
          acc[t][q] = __builtin_amdgcn_wmma_f32_16x16x4_f32(
              false, a, false, b[q], (short)0, acc[t][q], false, false);
      }
    }
  }

  // C/D layout: VGPR j -> row j (lanes 0-15) / row j+8 (lanes 16-31), col m.
  const int mb = (lane >> 4) << 3;
#pragma unroll
  for (int t = 0; t < RT; ++t) {
#pragma unroll
    for (int j = 0; j < 8; ++j) {
      const int orow = row0 + t * 16 + mb + j;
      if (orow < N) {
#pragma unroll
        for (int q = 0; q < COT; ++q)
          out[(size_t)orow * CO + co0 + q * 16 + m] = acc[t][q][j];
      }
    }
  }
}

// ---------------- sparse max pool (segment max via ordered-uint keys) -------
// Monotonic float->uint encoding so unsigned atomicMax implements float max.
__global__ void pool_init_kernel(unsigned* __restrict__ keys, int n) {
  const int i = blockIdx.x * blockDim.x + threadIdx.x;
  if (i < n) keys[i] = 0u;  // decodes below every real float
}

__global__ void pool_scatter_kernel(const float* __restrict__ x,
                                    const int* __restrict__ map,
                                    unsigned* __restrict__ keys, int N, int C) {
  const int i = blockIdx.x * blockDim.x + threadIdx.x;
  if (i >= N * C) return;
  const int n = i / C, c = i - n * C;
  const unsigned u   = __float_as_uint(x[i]);
  const unsigned key = (u & 0x80000000u) ? ~u : (u | 0x80000000u);
  atomicMax(&keys[(size_t)map[n] * C + c], key);
}

__global__ void pool_decode_kernel(const unsigned* __restrict__ keys,
                                   float* __restrict__ out, int n) {
  const int i = blockIdx.x * blockDim.x + threadIdx.x;
  if (i < n) {
    const unsigned key = keys[i];
    const unsigned u = (key & 0x80000000u) ? (key & 0x7fffffffu) : ~key;
    out[i] = __uint_as_float(u);
  }
}

// ---------------- host-side dispatch ---------------------------------------
static void launch_conv(const float* x, const int* nbr, const float* w,
                        float* out, int N, int Cin, int Co, hipStream_t s) {
  constexpr int RT  = 4;  // row tiles per wave
  constexpr int COT = 2;  // co tiles per wave (all CO are multiples of 32)
  dim3 grid((unsigned)((N + 16 * RT - 1) / (16 * RT)),
            (unsigned)(Co / (16 * COT)));
  dim3 block(32);
#define SUBM_CASE(CI, CO_)                                                     \
  if (Cin == CI && Co == CO_) {                                                \
    subm_conv_wmma<CI, CO_, RT, COT><<<grid, block, 0, s>>>(x, nbr, w, out, N);\
    return;                                                                    \
  }
  SUBM_CASE(3, 64)    SUBM_CASE(64, 64)   SUBM_CASE(64, 96)   SUBM_CASE(96, 96)
  SUBM_CASE(96, 128)  SUBM_CASE(128, 128) SUBM_CASE(128, 160) SUBM_CASE(160, 160)
  SUBM_CASE(160, 192) SUBM_CASE(192, 192) SUBM_CASE(192, 224) SUBM_CASE(224, 224)
  SUBM_CASE(224, 256) SUBM_CASE(256, 256)
#undef SUBM_CASE
}

extern "C" void kernel_launch(void* const* d_in, const int* in_sizes, int n_in,
                              void* d_out, int out_size, void* d_ws,
                              size_t ws_size, hipStream_t stream) {
  (void)n_in; (void)out_size; (void)ws_size;
  // Input order (setup_inputs dict order):
  //  0:nbr0 1:pool0 2:nbr1 3:pool1 4:nbr2 5:pool2 6:nbr3 7:pool3
  //  8:nbr4 9:pool4 10:nbr5 11:pool5 12:nbr6
  //  13..26: w0a w0b w1a w1b w2a w2b w3a w3b w4a w4b w5a w5b w6a w6b
  //  27:features 28:coors 29:batch_size
  static const int nbrIdx[7]  = {0, 2, 4, 6, 8, 10, 12};
  static const int poolIdx[6] = {1, 3, 5, 7, 9, 11};
  static const int C[7] = {64, 96, 128, 160, 192, 224, 256};

  int N[7];
  for (int l = 0; l < 7; ++l) N[l] = in_sizes[nbrIdx[l]] / KTAPS;

  size_t maxElems = 0;
  for (int l = 0; l < 7; ++l) {
    const size_t e = (size_t)N[l] * C[l];
    if (e > maxElems) maxElems = e;
  }

  float* bufA = (float*)d_ws;
  float* bufB = bufA + maxElems;
  unsigned* keys = (unsigned*)(bufB + maxElems);

  const float* cur = (const float*)d_in[27];  // features [N0, 3]
  float* p0 = bufA;
  float* p1 = bufB;

  for (int l = 0; l < 7; ++l) {
    const int* nbr   = (const int*)d_in[nbrIdx[l]];
    const float* wa  = (const float*)d_in[13 + 2 * l];
    const float* wb  = (const float*)d_in[13 + 2 * l + 1];
    const int Cin = (l == 0) ? 3 : C[l - 1];
    const int Cc  = C[l];

    launch_conv(cur, nbr, wa, p0, N[l], Cin, Cc, stream);
    float* bOut = (l == 6) ? (float*)d_out : p1;
    launch_conv(p0, nbr, wb, bOut, N[l], Cc, Cc, stream);

    if (l < 6) {
      const int* pm = (const int*)d_in[poolIdx[l]];
      const int outElems = N[l + 1] * Cc;
      const int inElems  = N[l] * Cc;
      const int thr = 256;
      pool_init_kernel<<<(outElems + thr - 1) / thr, thr, 0, stream>>>(keys, outElems);
      pool_scatter_kernel<<<(inElems + thr - 1) / thr, thr, 0, stream>>>(p1, pm, keys, N[l], Cc);
      pool_decode_kernel<<<(outElems + thr - 1) / thr, thr, 0, stream>>>(keys, p0, outElems);
      cur = p0;
      float* tmp = p0; p0 = p1; p1 = tmp;  // ping-pong
    }
  }
}